// Conv_77962246357097
// MI455X (gfx1250) — compile-verified
//
#include <hip/hip_runtime.h>
#include <hip/hip_bf16.h>

#define IMG_N   8
#define H       2048
#define W       2048
#define TILE_W  128
#define TILE_H  16
#define LDS_W   136            // 4 halo left + 128 + 4 slack right
#define LDS_H   18             // 1 halo top + 16 + 1 halo bottom
#define PADV    10.0f

typedef unsigned int u32x4 __attribute__((ext_vector_type(4)));
typedef int          i32x8 __attribute__((ext_vector_type(8)));
typedef int          i32x4 __attribute__((ext_vector_type(4)));

__global__ __launch_bounds__(256)
void local_min_stencil_tdm(const float* __restrict__ in, float* __restrict__ out)
{
    __shared__ __align__(16) float tile[LDS_H * LDS_W];

    const int t   = threadIdx.x;
    const int x0  = blockIdx.x * TILE_W;
    const int y0  = blockIdx.y * TILE_H;
    const int img = blockIdx.z;
    const size_t img_off = (size_t)img * (size_t)H * (size_t)W;
    const float* __restrict__ src = in + img_off;

    // Tile origin in global coords (clamped at top/left image edges).
    const int gx0  = (x0 == 0) ? 0 : (x0 - 4);
    const int gy0  = (y0 == 0) ? 0 : (y0 - 1);
    const int xoff = x0 - gx0;   // 0 or 4 : LDS col of x0
    const int yoff = y0 - gy0;   // 0 or 1 : LDS row of y0

    // ---- Wave 0 issues one Tensor-Data-Mover DMA for the whole tile --------
    if ((t >> 5) == 0) {
        const unsigned long long ga =
            (unsigned long long)(uintptr_t)(src + (size_t)gy0 * (size_t)W + (size_t)gx0);
        const unsigned lds_addr = (unsigned)(uintptr_t)(void*)&tile[0];
        const unsigned td0 = (unsigned)(W - gx0);   // remaining extent dim0 (OOB -> 0)
        const unsigned td1 = (unsigned)(H - gy0);   // remaining extent dim1 (OOB -> 0)

        u32x4 g0;
        g0[0] = 1u;                                             // count=1, user load
        g0[1] = lds_addr;                                       // lds_addr[31:0]
        g0[2] = (unsigned)ga;                                   // global_addr[31:0]
        g0[3] = (unsigned)((ga >> 32) & 0x1FFFFFFull)           // global_addr[56:32]
              | (2u << 30);                                     // type=2 ("image")

        i32x8 g1;
        g1[0] = (int)(2u << 16);                                // data_size=2 -> 4 bytes
        g1[1] = (int)((td0 & 0xFFFFu) << 16);                   // tensor_dim0[15:0]
        g1[2] = (int)((td0 >> 16) | ((td1 & 0xFFFFu) << 16));   // td0[31:16] | td1[15:0]
        g1[3] = (int)((td1 >> 16) | ((unsigned)LDS_W << 16));   // td1[31:16] | tile_dim0=136
        g1[4] = (int)LDS_H;                                     // tile_dim1=18, tile_dim2=0
        g1[5] = (int)W;                                         // tensor_dim0_stride = 2048
        g1[6] = 0;                                              // stride hi / dim1_stride lo
        g1[7] = 0;                                              // dim1_stride hi

        i32x4 gz  = {0, 0, 0, 0};                               // groups 2/3: unused (2D)
        i32x8 gz8 = {0, 0, 0, 0, 0, 0, 0, 0};                   // extended group: unused
        __builtin_amdgcn_tensor_load_to_lds(g0, g1, gz, gz, gz8, 0);
        __builtin_amdgcn_s_wait_tensorcnt(0);
    }
    __syncthreads();

    // ---- Compute: 2 rows x 4 cols per thread --------------------------------
    const int tx    = t & 31;           // 0..31 -> 4 columns each
    const int ty    = t >> 5;           // 0..7  -> rows ty, ty+8
    const int c0    = xoff + tx * 4;    // LDS column of first output pixel
    const int xbase = x0 + tx * 4;      // global column of first output pixel
    float* __restrict__ dst = out + img_off;

    #pragma unroll
    for (int rr = 0; rr < 2; ++rr) {
        const int ry = ty + rr * 8;     // 0..15 within tile
        const int y  = y0 + ry;         // global row
        const int r  = yoff + ry;       // LDS row of this pixel row

        const int rup = (r > 0) ? (r - 1) : 0;         // clamped (value unused if clamped)
        const int cl  = (c0 > 0) ? (c0 - 1) : 0;       // clamped (value unused if clamped)

        const float* rowc = &tile[r * LDS_W];
        float4 cen = *(const float4*)(rowc + c0);
        float  lft = rowc[cl];
        float  rgt = rowc[c0 + 4];
        float4 up  = *(const float4*)(&tile[rup * LDS_W] + c0);
        float4 dn  = *(const float4*)(&tile[(r + 1) * LDS_W] + c0);

        // Border padding (reference pads with 10.0)
        if (y == 0)         { up.x = PADV; up.y = PADV; up.z = PADV; up.w = PADV; }
        if (y == H - 1)     { dn.x = PADV; dn.y = PADV; dn.z = PADV; dn.w = PADV; }
        if (xbase == 0)       lft = PADV;   // left neighbor of x==0
        if (xbase + 4 == W)   rgt = PADV;   // right neighbor of x==W-1

        const float l0 = lft,   l1 = cen.x, l2 = cen.y, l3 = cen.z;
        const float r0 = cen.y, r1 = cen.z, r2 = cen.w, r3 = rgt;

        float4 res;
        res.x = (cen.x < up.x && cen.x < dn.x && cen.x < l0 && cen.x < r0) ? 1.0f : 0.0f;
        res.y = (cen.y < up.y && cen.y < dn.y && cen.y < l1 && cen.y < r1) ? 1.0f : 0.0f;
        res.z = (cen.z < up.z && cen.z < dn.z && cen.z < l2 && cen.z < r2) ? 1.0f : 0.0f;
        res.w = (cen.w < up.w && cen.w < dn.w && cen.w < l3 && cen.w < r3) ? 1.0f : 0.0f;

        *(float4*)(dst + (size_t)y * (size_t)W + (size_t)xbase) = res;
    }
}

extern "C" void kernel_launch(void* const* d_in, const int* in_sizes, int n_in,
                              void* d_out, int out_size, void* d_ws, size_t ws_size,
                              hipStream_t stream)
{
    (void)in_sizes; (void)n_in; (void)d_ws; (void)ws_size; (void)out_size;
    const float* data = (const float*)d_in[0];
    float* out = (float*)d_out;

    dim3 grid(W / TILE_W, H / TILE_H, IMG_N);   // (16, 128, 8)
    dim3 block(256);
    local_min_stencil_tdm<<<grid, block, 0, stream>>>(data, out);
}